// JPEGBlockingDetector_30674656428125
// MI455X (gfx1250) — compile-verified
//
#include <hip/hip_runtime.h>
#include <hip/hip_bf16.h>

#define HH 512
#define WW 512
#define NB 32

typedef float v2f __attribute__((ext_vector_type(2)));
typedef float v8f __attribute__((ext_vector_type(8)));

// ---------------- workspace layout (floats) ----------------
// [0,            16384)  : col edge-energy sums  (NB * WW)
// [16384,        32768)  : row edge-energy sums  (NB * HH)
// [32768,        33024)  : col phase flags       (NB * 8)
// [33024,        33280)  : row phase flags       (NB * 8)

__global__ __launch_bounds__(256) void zero_ws(float* ws, int n) {
    int i = blockIdx.x * 256 + threadIdx.x;
    if (i < n) ws[i] = 0.f;
}

// Fused single-pass reduction: each block owns a 32-row x 64-col lum tile
// (with +1 halo row/col), computes |dx| column partials and |dy| row partials
// from LDS, and atomically accumulates into global per-column / per-row sums.
__global__ __launch_bounds__(256) void reduce_kernel(const float* __restrict__ tgt,
                                                     float* __restrict__ wcol,
                                                     float* __restrict__ wrow) {
    __shared__ float lum[33][66];   // 65 used cols + pad to dodge bank conflicts
    __shared__ float red[256];
    const int t  = threadIdx.x;
    const int c0 = blockIdx.x * 64;
    const int r0 = blockIdx.y * 32;
    const int b  = blockIdx.z;
    const float* base = tgt + (size_t)b * 3 * HH * WW;

    for (int i = t; i < 33 * 65; i += 256) {
        int r = i / 65, c = i - r * 65;
        int gr = min(r0 + r, HH - 1);
        int gc = min(c0 + c, WW - 1);
        int off = gr * WW + gc;
        float R = base[off];
        float G = base[HH * WW + off];
        float Bv = base[2 * HH * WW + off];
        lum[r][c] = 0.299f * R + 0.587f * G + 0.114f * Bv;
    }
    __syncthreads();

    // column partials: 64 cols x 4 row-groups
    {
        int c = t & 63, rg = t >> 6;
        float acc = 0.f;
        for (int r = rg; r < 32; r += 4)
            acc += fabsf(lum[r][c] - lum[r][c + 1]);
        red[t] = acc;
        __syncthreads();
        if (rg == 0) {
            float s = red[c] + red[c + 64] + red[c + 128] + red[c + 192];
            int w = c0 + c;
            if (w < WW - 1) atomicAdd(&wcol[b * WW + w], s);
        }
        __syncthreads();
    }

    // row partials: 32 rows x 8 col-groups
    {
        int cg = t & 7, r = t >> 3;
        float acc = 0.f;
        for (int c = cg; c < 64; c += 8)
            acc += fabsf(lum[r][c] - lum[r + 1][c]);
        red[t] = acc;
        __syncthreads();
        if (cg == 0) {
            float s = 0.f;
            for (int j = 0; j < 8; ++j) s += red[t + j];   // t == r*8 here
            int h = r0 + r;
            if (h < HH - 1) atomicAdd(&wrow[b * HH + h], s);
        }
    }
}

// One wave32. psum[b][k] = sum_w line_sum[b][w] * onehot(w%8 == k), done as
// D(16x16) += A(16x4 line_sum chunk) x B(4x16 onehot chunk) with
// v_wmma_f32_16x16x4_f32. Then the NFA ratio test -> per-phase flags.
__global__ __launch_bounds__(32) void phase_kernel(const float* __restrict__ wcol,
                                                   const float* __restrict__ wrow,
                                                   float* __restrict__ fcol,
                                                   float* __restrict__ frow) {
    __shared__ float psum[2][NB][8];
    const int l = threadIdx.x;
    const int hf = l >> 4;      // lane half
    const int n  = l & 15;      // N column / M row index

    for (int dir = 0; dir < 2; ++dir) {
        const float* src = dir ? wrow : wcol;
        for (int mt = 0; mt < 2; ++mt) {
            const int b0 = mt * 16;
            v8f acc = {0.f, 0.f, 0.f, 0.f, 0.f, 0.f, 0.f, 0.f};
            for (int w0 = 0; w0 < 512; w0 += 4) {
                // A layout (ISA 7.12.2): lane m = l&15, VGPR j holds K = j + 2*half
                v2f a, bm;
                int abase = (b0 + n) * 512 + w0 + 2 * hf;
                a.x = src[abase + 0];
                a.y = src[abase + 1];
                // B (4x16 onehot): row k = j + 2*half, col n; line (w0+k) phase
                int k0 = (w0 + 2 * hf) & 7;
                int k1 = (w0 + 2 * hf + 1) & 7;
                bm.x = (k0 == n) ? 1.f : 0.f;
                bm.y = (k1 == n) ? 1.f : 0.f;
                acc = __builtin_amdgcn_wmma_f32_16x16x4_f32(
                    false, a, false, bm, (short)0, acc, false, false);
            }
            // C/D layout: VGPR v -> M = v (lanes 0-15) / v+8 (lanes 16-31)
            if (n < 8) {
                for (int v = 0; v < 8; ++v)
                    psum[dir][b0 + v + 8 * hf][n] = acc[v];
            }
        }
    }
    __syncthreads();

    // thread l -> batch l ; counts: 64 lines per phase except phase 7 -> 63
    for (int dir = 0; dir < 2; ++dir) {
        float tot = 0.f;
        for (int k = 0; k < 8; ++k) tot += psum[dir][l][k];
        float* dst = dir ? frow : fcol;
        for (int k = 0; k < 8; ++k) {
            float cnt = (k < 7) ? 64.f : 63.f;
            float oth = 511.f - cnt;
            float p = psum[dir][l][k];
            float a = p / (cnt * 512.f);
            float bg = (tot - p) / (oth * 512.f);
            dst[l * 8 + k] = (a / (bg + 1e-12f) > 100.f) ? 1.f : 0.f;
        }
    }
}

// Broadcast flags to the full (B,1,H,W) mask with 128-bit stores.
__global__ __launch_bounds__(256) void write_kernel(const float* __restrict__ fcol,
                                                    const float* __restrict__ frow,
                                                    float* __restrict__ out) {
    const int b = blockIdx.y;
    const int t = threadIdx.x;
    const int h = blockIdx.x * 2 + (t >> 7);
    const int w0 = (t & 127) * 4;

    float cf[8];
    for (int j = 0; j < 8; ++j) cf[j] = fcol[b * 8 + j];
    float rf = (h < HH - 1) ? frow[b * 8 + (h & 7)] : 0.f;
    bool rb = rf > 0.5f;

    float4 pix;
    float* p = (float*)&pix;
    for (int i = 0; i < 4; ++i) {
        int w = w0 + i;
        bool cb = (w < WW - 1) && (cf[w & 7] > 0.5f);
        p[i] = (cb || rb) ? 1.f : 0.f;
    }
    *(float4*)(out + (size_t)b * HH * WW + (size_t)h * WW + w0) = pix;
}

extern "C" void kernel_launch(void* const* d_in, const int* in_sizes, int n_in,
                              void* d_out, int out_size, void* d_ws, size_t ws_size,
                              hipStream_t stream) {
    const float* tgt = (const float*)d_in[0];
    float* out = (float*)d_out;
    float* ws = (float*)d_ws;

    float* wcol = ws;                 // NB*WW
    float* wrow = ws + NB * WW;       // NB*HH
    float* fcol = ws + 2 * NB * WW;   // NB*8
    float* frow = fcol + NB * 8;      // NB*8
    const int nzero = 2 * NB * WW + 2 * NB * 8;

    zero_ws<<<(nzero + 255) / 256, 256, 0, stream>>>(ws, nzero);
    reduce_kernel<<<dim3(WW / 64, HH / 32, NB), 256, 0, stream>>>(tgt, wcol, wrow);
    phase_kernel<<<1, 32, 0, stream>>>(wcol, wrow, fcol, frow);
    write_kernel<<<dim3(HH / 2, NB), 256, 0, stream>>>(fcol, frow, out);
}